// DGM_d_9887014716262
// MI455X (gfx1250) — compile-verified
//
#include <hip/hip_runtime.h>
#include <hip/hip_bf16.h>

typedef __attribute__((ext_vector_type(2))) float v2f;
typedef __attribute__((ext_vector_type(8))) float v8f;

#define BATCH 16
#define NN    2048
#define FIN   64
#define FOUT  32
#define KSEL  16
#define EPSQ  1e-8f
#define WAVES 8                       // waves per top-k block
#define MTILES (NN / 16)              // 128
#define STEPS  (MTILES / WAVES)       // 16

// ---------------------------------------------------------------------------
// Kernel 1: h = x @ W   via V_WMMA_F32_16X16X4_F32
// One wave computes one 16x16 tile of h. K = 64 -> 16 WMMAs.
// f32 A 16x4 layout: lane L: M = L&15; V0/V1 hold K = (L>>4)*2 + {0,1}.
// f32 B 4x16 layout: lane L: N = L&15; V0/V1 hold K = (L>>4)*2 + {0,1}.
// C/D 16x16: vgpr j: lanes 0-15 -> M=j, lanes 16-31 -> M=j+8; N = L&15.
// ---------------------------------------------------------------------------
__global__ __launch_bounds__(32) void embed_wmma(const float* __restrict__ x,
                                                 const float* __restrict__ W,
                                                 float* __restrict__ h) {
    int tile  = blockIdx.x;
    int mtile = tile >> 1;
    int ftile = tile & 1;
    int lane  = threadIdx.x;
    int r     = lane & 15;
    int khalf = (lane >> 4) * 2;
    int mbase = mtile * 16;
    int cbase = ftile * 16;

    v8f acc = {};
#pragma unroll
    for (int s = 0; s < FIN; s += 4) {
        int k0 = s + khalf;
        v2f a, b;
        a.x = x[(size_t)(mbase + r) * FIN + k0];
        a.y = x[(size_t)(mbase + r) * FIN + k0 + 1];
        b.x = W[(size_t)k0 * FOUT + cbase + r];
        b.y = W[(size_t)(k0 + 1) * FOUT + cbase + r];
        acc = __builtin_amdgcn_wmma_f32_16x16x4_f32(false, a, false, b,
                                                    (short)0, acc, false, false);
    }
    int rowoff = (lane >> 4) * 8;
#pragma unroll
    for (int j = 0; j < 8; ++j) {
        h[(size_t)(mbase + rowoff + j) * FOUT + cbase + r] = acc[j];
    }
}

// ---------------------------------------------------------------------------
// Kernel 2: sq[i] = sum_j h[i][j]^2
// ---------------------------------------------------------------------------
__global__ void sq_kernel(const float* __restrict__ h, float* __restrict__ sq,
                          int total) {
    int i = blockIdx.x * blockDim.x + threadIdx.x;
    if (i >= total) return;
    float s = 0.f;
#pragma unroll
    for (int j = 0; j < FOUT; ++j) {
        float v = h[(size_t)i * FOUT + j];
        s += v * v;
    }
    sq[i] = s;
}

// ---------------------------------------------------------------------------
// Kernel 3: one 256-thread block (8 waves) per (batch, 16-row tile).
// Wave w handles m-tiles {t*8 + w}: Gram tile via 8x WMMA f32 16x16x4 (K=32),
// D = max(0, sq_n + sq_m - 2G), lq = D*expT - log(-log(q+eps)).
// Each wave keeps a private per-row top-16 list in LDS; wave 0 merges the
// 8x16 candidates per row at the end and emits sorted logprobs + edges.
// 8 waves/block * 2048 blocks = 16K waves to hide HBM latency on the q stream
// (q = 268 MB is the roofline term: ~11.5 us at 23.3 TB/s).
// ---------------------------------------------------------------------------
__global__ __launch_bounds__(256) void topk_kernel(const float* __restrict__ h,
                                                   const float* __restrict__ sq,
                                                   const float* __restrict__ q,
                                                   const float* __restrict__ temperature,
                                                   float* __restrict__ logprobs,
                                                   int* __restrict__ edges) {
    __shared__ float stage[WAVES][16][17];
    __shared__ float lv[WAVES][16][KSEL];
    __shared__ int   li[WAVES][16][KSEL];

    int blk   = blockIdx.x;
    int b     = blk >> 7;            // / (N/16) = /128
    int ntile = blk & 127;
    int nbase = ntile * 16;
    int tid   = threadIdx.x;
    int w     = tid >> 5;            // wave id 0..7
    int lane  = tid & 31;
    int r     = lane & 15;
    int half  = lane >> 4;
    int khalf = half * 2;

    float t0 = temperature[0];
    t0 = fminf(fmaxf(t0, -5.f), 5.f);
    float expT = expf(t0);

    const float* hb  = h  + (size_t)b * NN * FOUT;
    const float* sqb = sq + (size_t)b * NN;
    const float* qb  = q  + (size_t)b * NN * NN;

    // init this wave's per-row top-k lists
    if (lane < 16) {
#pragma unroll
        for (int j = 0; j < KSEL; ++j) {
            lv[w][lane][j] = __builtin_inff();
            li[w][lane][j] = 0;
        }
    }
    float worst = __builtin_inff();
    int   wslot = 0;

    // register-resident A fragment: rows nbase..nbase+15, K = 0..31
    v2f afrag[8];
#pragma unroll
    for (int s = 0; s < 8; ++s) {
        int k0 = 4 * s + khalf;
        afrag[s].x = hb[(size_t)(nbase + r) * FOUT + k0];
        afrag[s].y = hb[(size_t)(nbase + r) * FOUT + k0 + 1];
    }
    float sqn[8];
#pragma unroll
    for (int j = 0; j < 8; ++j) sqn[j] = sqb[nbase + half * 8 + j];

    __syncthreads();

    for (int t = 0; t < STEPS; ++t) {
        int mtile = t * WAVES + w;
        int mbase = mtile * 16;

        // prefetch this wave's next q tile (gfx1250 global_prefetch_b8 path)
        if (t + 1 < STEPS) {
            const float* pq = &qb[(size_t)(nbase + half * 8) * NN + mbase + WAVES * 16];
            __builtin_prefetch(pq, 0, 0);
        }

        v8f acc = {};
#pragma unroll
        for (int s = 0; s < 8; ++s) {
            int k0 = 4 * s + khalf;
            v2f bfrag;
            bfrag.x = hb[(size_t)(mbase + r) * FOUT + k0];
            bfrag.y = hb[(size_t)(mbase + r) * FOUT + k0 + 1];
            acc = __builtin_amdgcn_wmma_f32_16x16x4_f32(false, afrag[s], false, bfrag,
                                                        (short)0, acc, false, false);
        }

        float sqm = sqb[mbase + r];   // column index = r
#pragma unroll
        for (int j = 0; j < 8; ++j) {
            int row = half * 8 + j;
            float D = sqn[j] + sqm - 2.0f * acc[j];
            D = fmaxf(D, 0.f);
            float qv = qb[(size_t)(nbase + row) * NN + mbase + r];
            float g  = logf(-logf(qv + EPSQ));
            stage[w][row][r] = D * expT - g;
        }
        __syncthreads();   // uniform across all waves (same STEPS everywhere)

        if (lane < 16) {
            // lane r owns row r within this wave's list
            for (int c = 0; c < 16; ++c) {
                float v = stage[w][r][c];
                if (v < worst) {
                    lv[w][r][wslot] = v;
                    li[w][r][wslot] = mbase + c;
                    worst = lv[w][r][0]; wslot = 0;
#pragma unroll
                    for (int j = 1; j < KSEL; ++j) {
                        float tv = lv[w][r][j];
                        if (tv > worst) { worst = tv; wslot = j; }
                    }
                }
            }
        }
        __syncthreads();
    }

    // merge: wave 0, lane r selects global top-16 (ascending lq) for row r
    // out of the 8 per-wave lists (128 candidates), destructively.
    if (w == 0 && lane < 16) {
        int n = nbase + r;
        size_t base = ((size_t)b * NN + n) * KSEL;
        size_t erow = (size_t)BATCH * NN * KSEL;
        for (int j = 0; j < KSEL; ++j) {
            float mval = __builtin_inff();
            int   mw = 0, ms = 0;
#pragma unroll
            for (int wv = 0; wv < WAVES; ++wv) {
                for (int s2 = 0; s2 < KSEL; ++s2) {
                    float tv = lv[wv][r][s2];
                    if (tv < mval) { mval = tv; mw = wv; ms = s2; }
                }
            }
            lv[mw][r][ms] = __builtin_inff();
            int mi = li[mw][r][ms];

            logprobs[base + j]     = -mval;          // top_k(-lq), descending
            edges[base + j]        = mi + b * NN;    // row 0: neighbor idx + b*N
            edges[erow + base + j] = n  + b * NN;    // row 1: source row + b*N
        }
    }
}

// ---------------------------------------------------------------------------
extern "C" void kernel_launch(void* const* d_in, const int* in_sizes, int n_in,
                              void* d_out, int out_size, void* d_ws, size_t ws_size,
                              hipStream_t stream) {
    const float* x           = (const float*)d_in[0];
    const float* W           = (const float*)d_in[1];
    const float* temperature = (const float*)d_in[2];
    const float* q           = (const float*)d_in[3];
    // d_in[4] = k (fixed 16)

    float* h        = (float*)d_out;                                   // B*N*FOUT f32
    int*   edges    = (int*)(h + (size_t)BATCH * NN * FOUT);           // 2*B*N*k  i32
    float* logprobs = (float*)(edges + (size_t)2 * BATCH * NN * KSEL); // B*N*k    f32
    float* sqw      = (float*)d_ws;                                    // B*N      f32

    embed_wmma<<<(BATCH * NN / 16) * (FOUT / 16), 32, 0, stream>>>(x, W, h);
    sq_kernel<<<(BATCH * NN + 255) / 256, 256, 0, stream>>>(h, sqw, BATCH * NN);
    topk_kernel<<<BATCH * (NN / 16), 256, 0, stream>>>(h, sqw, q, temperature,
                                                       logprobs, edges);
}